// GCN_38628935860365
// MI455X (gfx1250) — compile-verified
//
#include <hip/hip_runtime.h>

typedef __attribute__((ext_vector_type(2))) float v2f;
typedef __attribute__((ext_vector_type(8))) float v8f;

#define N_NODES   100000
#define N_FEAT    512
#define N_HID     256
#define N_EDGES   3200000
// LDS row stride padded by 4 floats so lanes m=0..15 hit banks 4m, 4m+1 (no conflicts)
#define XS_STRIDE (N_FEAT + 4)

// ---------------------------------------------------------------------------
// Zero the aggregation buffer (d_out) each launch (harness poisons it).
// ---------------------------------------------------------------------------
__global__ __launch_bounds__(256) void zero_f32_k(float4* __restrict__ p, int n4) {
    int i = blockIdx.x * 256 + threadIdx.x;
    if (i < n4) p[i] = make_float4(0.f, 0.f, 0.f, 0.f);
}

// ---------------------------------------------------------------------------
// h = x @ W^T + b   via V_WMMA_F32_16X16X4_F32  (exact f32 matrix path)
//   block = 256 threads (8 waves), computes a 16-row x 256-col strip of h.
//   x strip (16x512) staged in LDS once, shared by all 8 waves.
//   wave w computes col tiles [32w,32w+16) and [32w+16,32w+32).
// A-layout (32-bit A 16x4): lane l -> row m=l&15, K-pair koff=(l>>4)*2.
// B-layout (32-bit B 4x16): lane l -> col n=l&15, same K-pair.
// C/D layout: VGPR r, lane l -> row (l>>4)*8 + r, col l&15.
// ---------------------------------------------------------------------------
__global__ __launch_bounds__(256) void gemm_wmma_k(const float* __restrict__ x,
                                                   const float* __restrict__ W,
                                                   const float* __restrict__ bias,
                                                   float* __restrict__ h) {
    __shared__ float xs[16 * XS_STRIDE];   // 33 KB

    const int row0 = blockIdx.x * 16;      // N_NODES % 16 == 0 -> no tail
    const int tid  = threadIdx.x;

    // Cooperative stage of the 16x512 x-strip into LDS (float4 per op).
    const float4* xg = (const float4*)(x + (size_t)row0 * N_FEAT);
    for (int i = tid; i < 16 * (N_FEAT / 4); i += 256) {
        int r    = i >> 7;        // row in strip (128 float4 per row)
        int coff = i & 127;       // float4 within row
        *(float4*)(&xs[r * XS_STRIDE + coff * 4]) = xg[r * (N_FEAT / 4) + coff];
    }
    __syncthreads();

    const int wave = tid >> 5;             // 0..7
    const int lane = tid & 31;
    const int col0 = wave * 32;            // two 16-wide col tiles
    const int m    = lane & 15;
    const int koff = (lane >> 4) * 2;

    const float* xp  = xs + m * XS_STRIDE + koff;
    const float* Wp0 = W + (size_t)(col0 + m)      * N_FEAT + koff;
    const float* Wp1 = W + (size_t)(col0 + 16 + m) * N_FEAT + koff;

    v8f c0 = {};
    v8f c1 = {};
#pragma unroll 4
    for (int k = 0; k < N_FEAT; k += 4) {
        v2f a  = *(const v2f*)(xp  + k);   // ds_load_b64
        v2f b0 = *(const v2f*)(Wp0 + k);   // global_load_b64 (W resident in L2)
        v2f b1 = *(const v2f*)(Wp1 + k);
        c0 = __builtin_amdgcn_wmma_f32_16x16x4_f32(false, a, false, b0,
                                                   (short)0, c0, false, false);
        c1 = __builtin_amdgcn_wmma_f32_16x16x4_f32(false, a, false, b1,
                                                   (short)0, c1, false, false);
    }

    // Epilogue: add bias, store. Coalesced across lanes 0..15 / 16..31 per r.
    const int rbase = (lane >> 4) * 8;
    const int n0 = col0 + m;
    const int n1 = col0 + 16 + m;
    const float bb0 = bias[n0];
    const float bb1 = bias[n1];
#pragma unroll
    for (int r = 0; r < 8; ++r) {
        const size_t rr = (size_t)(row0 + rbase + r) * N_HID;
        h[rr + n0] = c0[r] + bb0;
        h[rr + n1] = c1[r] + bb1;
    }
}

// ---------------------------------------------------------------------------
// SpMM scatter: one wave per edge. Gather h[col] (2 x b128 per lane), scale,
// atomic-add into out[row]. h and out both fit in the 192 MB L2, so these
// gathers/atomics run at L2 bandwidth, not HBM.
// ---------------------------------------------------------------------------
__global__ __launch_bounds__(256) void spmm_scatter_k(const int*   __restrict__ rows,
                                                      const int*   __restrict__ cols,
                                                      const float* __restrict__ vals,
                                                      const float* __restrict__ h,
                                                      float*       __restrict__ out) {
    const int wid = blockIdx.x * 8 + (threadIdx.x >> 5);
    if (wid >= N_EDGES) return;
    const int   lane = threadIdx.x & 31;
    const int   r = rows[wid];          // wave-uniform -> scalar loads
    const int   c = cols[wid];
    const float v = vals[wid];

    const float4* src = (const float4*)(h + (size_t)c * N_HID);
    float*        dst = out + (size_t)r * N_HID;
#pragma unroll
    for (int i = 0; i < 2; ++i) {
        const int j = lane + i * 32;    // float4 index 0..63 (256 floats)
        float4 mv = src[j];
        atomicAdd(dst + j * 4 + 0, v * mv.x);
        atomicAdd(dst + j * 4 + 1, v * mv.y);
        atomicAdd(dst + j * 4 + 2, v * mv.z);
        atomicAdd(dst + j * 4 + 3, v * mv.w);
    }
}

// ---------------------------------------------------------------------------
// PReLU in place on the aggregated output.
// ---------------------------------------------------------------------------
__global__ __launch_bounds__(256) void prelu_k(float4* __restrict__ p,
                                               const float* __restrict__ alpha,
                                               int n4) {
    int i = blockIdx.x * 256 + threadIdx.x;
    if (i >= n4) return;
    const float a = alpha[0];
    float4 v = p[i];
    v.x = v.x >= 0.f ? v.x : a * v.x;
    v.y = v.y >= 0.f ? v.y : a * v.y;
    v.z = v.z >= 0.f ? v.z : a * v.z;
    v.w = v.w >= 0.f ? v.w : a * v.w;
    p[i] = v;
}

// ---------------------------------------------------------------------------
extern "C" void kernel_launch(void* const* d_in, const int* in_sizes, int n_in,
                              void* d_out, int out_size, void* d_ws, size_t ws_size,
                              hipStream_t stream) {
    const float* x        = (const float*)d_in[0];   // [1, N, F]
    const int*   adj_rows = (const int*)  d_in[1];   // [E]
    const int*   adj_cols = (const int*)  d_in[2];   // [E]
    const float* adj_vals = (const float*)d_in[3];   // [E]
    const float* W        = (const float*)d_in[4];   // [H, F]
    const float* b        = (const float*)d_in[5];   // [H]
    const float* alpha    = (const float*)d_in[6];   // [1]
    float* out = (float*)d_out;                      // [1, N, H]
    float* h   = (float*)d_ws;                       // N*H floats = 102.4 MB scratch

    const int n4 = N_NODES * N_HID / 4;              // 6.4M float4

    zero_f32_k<<<(n4 + 255) / 256, 256, 0, stream>>>((float4*)out, n4);
    gemm_wmma_k<<<N_NODES / 16, 256, 0, stream>>>(x, W, b, h);
    spmm_scatter_k<<<(N_EDGES + 7) / 8, 256, 0, stream>>>(adj_rows, adj_cols,
                                                          adj_vals, h, out);
    prelu_k<<<(n4 + 255) / 256, 256, 0, stream>>>((float4*)out, alpha, n4);
}